// Sentimental_Single_Rnn_53085795779397
// MI455X (gfx1250) — compile-verified
//
#include <hip/hip_runtime.h>
#include <math.h>

// Problem constants (match reference)
#define BSZ   256
#define TLEN  1024
#define VOCAB 5000
#define DIM   100
#define HD    64
#define CC    2

typedef __attribute__((ext_vector_type(2))) float v2f;
typedef __attribute__((ext_vector_type(8))) float v8f;

// Branch-free tanh: hardware V_TANH_F32 (confirmed lowering on gfx1250),
// else exp2/rcp identity.
static __device__ __forceinline__ float fast_tanhf(float x) {
#if __has_builtin(__builtin_amdgcn_tanhf)
  return __builtin_amdgcn_tanhf(x);
#elif __has_builtin(__builtin_amdgcn_tanh_f32)
  return __builtin_amdgcn_tanh_f32(x);
#else
  float ax = __builtin_fabsf(x);
  float e = __builtin_amdgcn_exp2f(ax * -2.885390081777927f);   // exp(-2ax)
  float r = __builtin_amdgcn_rcpf(1.0f + e);
  float t = __builtin_fmaf(-2.0f * e, r, 1.0f);                 // (1-e)/(1+e)
  return __builtin_copysignf(t, x);
#endif
}

// ---------------------------------------------------------------------------
// Kernel 1: EW[v][h] = sum_d E[v][d] * W[d][h] + b[h]   (5000 x 64, 64 MFLOP)
// ---------------------------------------------------------------------------
__global__ void ew_bias_kernel(const float* __restrict__ E,
                               const float* __restrict__ W,
                               const float* __restrict__ b,
                               float* __restrict__ EW) {
  int idx = blockIdx.x * blockDim.x + threadIdx.x;
  if (idx >= VOCAB * HD) return;
  int v = idx >> 6;
  int h = idx & 63;
  const float* Er = E + v * DIM;
  float acc = b[h];
#pragma unroll 4
  for (int d = 0; d < DIM; ++d) acc = fmaf(Er[d], W[d * HD + h], acc);
  EW[idx] = acc;
}

// ---------------------------------------------------------------------------
// Kernel 2: transpose tokens [B,T] -> [T,B]
// ---------------------------------------------------------------------------
__global__ void tok_transpose_kernel(const int* __restrict__ tokens,
                                     int* __restrict__ tokT) {
  int idx = blockIdx.x * blockDim.x + threadIdx.x;
  if (idx >= BSZ * TLEN) return;
  int b = idx / TLEN;
  int t = idx - b * TLEN;
  tokT[t * BSZ + b] = tokens[idx];
}

// ---------------------------------------------------------------------------
// Kernel 3: sequential masked RNN scan, fused time-mean + sigmoid head.
// 16 blocks x 128 threads (4 waves); wave nt owns output cols [nt*16,nt*16+16).
// Per step: A-fragments of h bulk-preloaded from LDS (one clustered batch of
// ds_load_2addr_b64), then h@U as two independent 8-deep chains of
// V_WMMA_F32_16X16X4_F32; U resident in registers; h ping-pongs between two
// LDS tiles (one barrier/step); tokens prefetched a step ahead so the 8 EW
// gathers issue at loop top under the WMMA chains; carry h in registers.
// ---------------------------------------------------------------------------
__global__ void __launch_bounds__(128)
rnn_scan_kernel(const int* __restrict__ tokT, const float* __restrict__ EW,
                const float* __restrict__ U, const float* __restrict__ Wd,
                const float* __restrict__ bd, float* __restrict__ out) {
  __shared__ float hbuf[2][16 * HD];   // ping-pong h tiles [row][k]

  const int lane   = threadIdx.x & 31;
  const int nt     = threadIdx.x >> 5;       // N-tile of this wave (0..3)
  const int base_b = blockIdx.x * 16;        // first batch row of this block
  const int kb     = (lane < 16) ? 0 : 2;    // K sub-pair per lane half
  const int nl     = lane & 15;
  const int n      = nt * 16 + nl;           // owned output column (D layout)
  const int mlo    = (lane < 16) ? 0 : 8;    // row base for D elements

  // Preload U as 16 B-fragments (K-chunks of 4): 32 VGPRs, resident all steps.
  v2f ufrag[16];
#pragma unroll
  for (int kc = 0; kc < 16; ++kc) {
    ufrag[kc].x = U[(kc * 4 + kb + 0) * HD + n];
    ufrag[kc].y = U[(kc * 4 + kb + 1) * HD + n];
  }

  // h0 = 0 in ping buffer 0; carry + mean accumulators in registers.
  for (int i = threadIdx.x; i < 16 * HD; i += 128) hbuf[0][i] = 0.0f;
  v8f hprev = {};
  v8f hsum  = {};

  // Prefetch tokens for t = 0 (contiguous 8 ints -> b128 loads).
  int tokv[8];
#pragma unroll
  for (int r = 0; r < 8; ++r) tokv[r] = tokT[base_b + mlo + r];

  __syncthreads();

  int p = 0;
  for (int t = 0; t < TLEN; ++t) {
    // Issue all 8 xW gathers immediately (tokens already in registers);
    // latency overlaps the A-fragment loads and WMMA chains below.
    float xw[8];
#pragma unroll
    for (int r = 0; r < 8; ++r) xw[r] = EW[tokv[r] * HD + n];

    // Bulk-preload all 16 A-fragments of h_prev: 8 clustered
    // ds_load_2addr_b64, so DS latency pipelines under the WMMA chains.
    const float* hp = hbuf[p];
    v2f afr[16];
#pragma unroll
    for (int kc = 0; kc < 16; ++kc)
      afr[kc] = *(const v2f*)&hp[nl * HD + kc * 4 + kb];

    // acc = h_prev @ U : two independent 8-deep WMMA chains.
    v8f acc0 = {};
    v8f acc1 = {};
#pragma unroll
    for (int kc = 0; kc < 8; ++kc) {
      acc0 = __builtin_amdgcn_wmma_f32_16x16x4_f32(
          false, afr[kc], false, ufrag[kc], (short)0, acc0, false, false);
      acc1 = __builtin_amdgcn_wmma_f32_16x16x4_f32(
          false, afr[kc + 8], false, ufrag[kc + 8], (short)0, acc1, false, false);
    }

    // Prefetch tokens for t+1 (clamped; scalar select, no branch).
    const int  tn  = (t + 1 < TLEN) ? (t + 1) : t;
    const int* tkn = tokT + tn * BSZ + base_b;
    int toknext[8];
#pragma unroll
    for (int r = 0; r < 8; ++r) toknext[r] = tkn[mlo + r];

    v8f acc = acc0 + acc1;   // v_pk_add_f32 x4

    // Masked tanh update; write new h into the pong buffer.
    float* hw = hbuf[p ^ 1];
#pragma unroll
    for (int r = 0; r < 8; ++r) {
      float hn = fast_tanhf(xw[r] + acc[r]);
      float hv = (tokv[r] != 0) ? hn : hprev[r];     // Keras masked carry
      hprev[r] = hv;
      hw[(mlo + r) * HD + n] = hv;
      hsum[r] += hv;                                 // unmasked time-mean accum
    }
#pragma unroll
    for (int r = 0; r < 8; ++r) tokv[r] = toknext[r];

    __syncthreads();   // pong writes visible; ping safe to overwrite next iter
    p ^= 1;
  }

  // pooled head: stage hsum rows in LDS (reuse buffer 0), then 16x2 outputs
#pragma unroll
  for (int r = 0; r < 8; ++r) hbuf[0][(mlo + r) * HD + n] = hsum[r];
  __syncthreads();

  if (threadIdx.x < 32) {
    int row = (int)threadIdx.x >> 1;
    int c   = (int)threadIdx.x & 1;
    float acc2 = 0.0f;
#pragma unroll 8
    for (int k = 0; k < HD; ++k)
      acc2 = fmaf(hbuf[0][row * HD + k], Wd[k * CC + c], acc2);
    float logit = acc2 * (1.0f / (float)TLEN) + bd[c];  // mean folded into dot
    float e = __builtin_amdgcn_exp2f(logit * -1.4426950408889634f);
    out[(base_b + row) * CC + c] = __builtin_amdgcn_rcpf(1.0f + e);
  }
}

// ---------------------------------------------------------------------------
extern "C" void kernel_launch(void* const* d_in, const int* in_sizes, int n_in,
                              void* d_out, int out_size, void* d_ws, size_t ws_size,
                              hipStream_t stream) {
  const int*   tokens = (const int*)d_in[0];
  const float* E      = (const float*)d_in[1];
  const float* W      = (const float*)d_in[2];
  const float* U      = (const float*)d_in[3];
  const float* b      = (const float*)d_in[4];
  const float* Wd     = (const float*)d_in[5];
  const float* bd     = (const float*)d_in[6];
  float* out = (float*)d_out;

  // Workspace: EW (5000*64 f32 = 1.28 MB) | tokensT (1024*256 i32 = 1 MB)
  float* EW   = (float*)d_ws;
  int*   tokT = (int*)((char*)d_ws + (size_t)VOCAB * HD * sizeof(float));

  ew_bias_kernel<<<(VOCAB * HD + 255) / 256, 256, 0, stream>>>(E, W, b, EW);
  tok_transpose_kernel<<<(BSZ * TLEN + 255) / 256, 256, 0, stream>>>(tokens, tokT);
  rnn_scan_kernel<<<BSZ / 16, 128, 0, stream>>>(tokT, EW, U, Wd, bd, out);
}